// TypeGraphConvolution_14139032338678
// MI455X (gfx1250) — compile-verified
//
#include <hip/hip_runtime.h>
#include <hip/hip_bf16.h>

// Problem constants (match reference): B=4, L=256, D_IN=D_OUT=256
#define NB 4
#define NL 256
#define ND 256
#define JS 8            // j-reduction segments (parallelism for the HBM stream)
#define JCH (NL / JS)   // 32 j's per segment

typedef __attribute__((ext_vector_type(2))) float v2f;
typedef __attribute__((ext_vector_type(4))) float v4f;
typedef __attribute__((ext_vector_type(8))) float v8f;

// ---------------------------------------------------------------------------
// Kernel 1: pure dep_embed stream (the 268 MB / ~11.5us roofline term).
//   Epart[seg][b,i,d] = sum_{j in seg} adj[b,i,j] * dep[b,j,i,d]
//   Spart[seg][b,i]   = sum_{j in seg} adj[b,i,j]
// Block = 256 threads = 4 "i" rows x 64 d-quads (float4/lane).
// Grid = (B*L/4, JS) = 2048 blocks = 16K waves -> enough in-flight b128 loads
// to cover HBM latency at 23.3 TB/s. dep loads are NON-TEMPORAL so the
// single-use 268 MB stream doesn't thrash the 192 MB L2.
// Hot loop: 1 NT b128 load + 4 FMAs + 1 LDS broadcast per iteration.
// ---------------------------------------------------------------------------
__global__ __launch_bounds__(256) void tgc_dep_reduce_kernel(
    const float* __restrict__ adj, const float* __restrict__ dep,
    float* __restrict__ Epart, float* __restrict__ Spart)
{
    __shared__ float adjS[4 * JCH];

    const int t   = threadIdx.x;
    const int sub = t >> 6;            // which of 4 i-rows in this block
    const int tt  = t & 63;            // d-quad index (d = 4*tt)
    const int g0  = blockIdx.x << 2;   // first global row (b*L + i)
    const int b   = g0 >> 8;
    const int i0  = g0 & 255;
    const int seg = blockIdx.y;
    const int j0  = seg * JCH;

    // Stage the 4x32 adj slice for this block's rows / j-segment
    if (t < 4 * JCH) {
        const int r = t >> 5;          // row within block (JCH == 32)
        const int c = t & 31;          // j within segment
        adjS[t] = adj[((size_t)b * NL + (i0 + r)) * NL + j0 + c];
    }
    __syncthreads();

    const int i = i0 + sub;
    const float* dp = dep + (((size_t)b * NL + j0) * NL + i) * ND + 4 * tt;

    v4f  acc  = {0.f, 0.f, 0.f, 0.f};
    float sadj = 0.f;

#pragma unroll 8
    for (int j = 0; j < JCH; ++j) {
        const float a  = adjS[sub * JCH + j];
        const v4f   dv = __builtin_nontemporal_load(
                             (const v4f*)(dp + (size_t)j * (NL * ND)));
        acc  += dv * a;
        sadj += a;
    }

    const size_t m = (size_t)b * NL + i;
    *(v4f*)(Epart + ((size_t)seg * (NB * NL) + m) * ND + 4 * tt) = acc;
    if (tt == 0) Spart[(size_t)seg * (NB * NL) + m] = sadj;
}

// ---------------------------------------------------------------------------
// Kernel 2: H[b,i,d] = sum_j adj[b,i,j] * text[b,j,d] via WMMA (replaces the
// 256 MB cached text re-read of the naive fusion with a 134 MFLOP GEMM), then
// G = H + sum_seg Epart.  Per-batch GEMM: M=N=K=256, fp32.
// One wave per 16x16 tile, V_WMMA_F32_16X16X4_F32, lane = 16h + idx:
//   A frag = { adj[b, m0+idx, k+2h], adj[b, m0+idx, k+2h+1] }   (contiguous)
//   B frag = { text[b, k+2h, n0+idx], text[b, k+2h+1, n0+idx] } (K-strided)
//   C/D:  VGPR r -> row m0 + r + 8h, col n0 + idx.
// Grid = (B, 16, 16); exact tiling -> EXEC all-1s (WMMA requirement).
// ---------------------------------------------------------------------------
__global__ __launch_bounds__(32) void tgc_text_wmma_kernel(
    const float* __restrict__ adj, const float* __restrict__ text,
    const float* __restrict__ Epart, float* __restrict__ G)
{
    const int lane = threadIdx.x & 31;
    const int h    = lane >> 4;
    const int idx  = lane & 15;
    const int b    = blockIdx.x;
    const int m0   = blockIdx.y * 16;   // i tile
    const int n0   = blockIdx.z * 16;   // d tile

    const float* aRow = adj  + ((size_t)b * NL + m0 + idx) * NL; // adj[b,i,:]
    const float* tCol = text + (size_t)b * NL * ND + n0 + idx;   // text[b,k,n]

    v8f c = {};
#pragma unroll 8
    for (int k = 0; k < NL; k += 4) {
        const int kk = k + 2 * h;
        const v2f a = *(const v2f*)(aRow + kk);
        v2f bb;
        bb.x = tCol[(size_t)kk * ND];
        bb.y = tCol[(size_t)(kk + 1) * ND];
        c = __builtin_amdgcn_wmma_f32_16x16x4_f32(
                false, a, false, bb, (short)0, c, false, false);
    }

#pragma unroll
    for (int r = 0; r < 8; ++r) {
        const size_t m   = (size_t)b * NL + m0 + r + 8 * h;
        const size_t off = m * ND + n0 + idx;
        float g = c[r];
#pragma unroll
        for (int s = 0; s < JS; ++s)
            g += Epart[(size_t)s * (NB * NL * ND) + off];
        G[off] = g;
    }
}

// ---------------------------------------------------------------------------
// Kernel 3: out[m,o] = sum_d G[m,d] * W[o,d] + S[m] * bias[o]
// (M=1024, N=256, K=256 fp32 WMMA; S[m] reduced from Spart in the epilogue.)
//   A frag = { G[m0+idx, k+2h], G[m0+idx, k+2h+1] }   (contiguous)
//   B frag = { W[n0+idx, k+2h], W[n0+idx, k+2h+1] }   (contiguous: W is [o,d])
// Grid = (64, 16) waves; exact tiling -> EXEC all-1s.
// ---------------------------------------------------------------------------
__global__ __launch_bounds__(32) void tgc_out_wmma_kernel(
    const float* __restrict__ G, const float* __restrict__ Wm,
    const float* __restrict__ bias, const float* __restrict__ Spart,
    float* __restrict__ out)
{
    const int lane = threadIdx.x & 31;
    const int h    = lane >> 4;
    const int idx  = lane & 15;
    const int m0   = blockIdx.x * 16;   // row tile (b*L + i)
    const int n0   = blockIdx.y * 16;   // o tile

    const float* aRow = G  + (size_t)(m0 + idx) * ND;
    const float* bRow = Wm + (size_t)(n0 + idx) * ND;

    v8f c = {};
#pragma unroll 8
    for (int k = 0; k < ND; k += 4) {
        const v2f a  = *(const v2f*)(aRow + k + 2 * h);
        const v2f bb = *(const v2f*)(bRow + k + 2 * h);
        c = __builtin_amdgcn_wmma_f32_16x16x4_f32(
                false, a, false, bb, (short)0, c, false, false);
    }

    const float bv = bias[n0 + idx];
#pragma unroll
    for (int r = 0; r < 8; ++r) {
        const int row = m0 + r + 8 * h;
        float s = 0.f;
#pragma unroll
        for (int sg = 0; sg < JS; ++sg)
            s += Spart[(size_t)sg * (NB * NL) + row];
        out[(size_t)row * ND + n0 + idx] = c[r] + s * bv;
    }
}

// ---------------------------------------------------------------------------
// Inputs (setup_inputs order): text[B,L,D], adj[B,L,L], dep_embed[B,L,L,D],
//                              W[D_OUT,D_IN], b[D_OUT]. Output: [B,L,D_OUT] f32.
// Workspace: Epart 8 MB + G 1 MB + Spart 32 KB.
// ---------------------------------------------------------------------------
extern "C" void kernel_launch(void* const* d_in, const int* in_sizes, int n_in,
                              void* d_out, int out_size, void* d_ws, size_t ws_size,
                              hipStream_t stream) {
    (void)in_sizes; (void)n_in; (void)out_size; (void)ws_size;

    const float* text = (const float*)d_in[0];
    const float* adj  = (const float*)d_in[1];
    const float* dep  = (const float*)d_in[2];
    const float* Wm   = (const float*)d_in[3];
    const float* bias = (const float*)d_in[4];
    float* out = (float*)d_out;

    float* Epart = (float*)d_ws;                          // [JS][B*L, D]
    float* G     = Epart + (size_t)JS * NB * NL * ND;     // [B*L, D]
    float* Spart = G + (size_t)NB * NL * ND;              // [JS][B*L]

    tgc_dep_reduce_kernel<<<dim3(NB * NL / 4, JS), dim3(256), 0, stream>>>(
        adj, dep, Epart, Spart);
    tgc_text_wmma_kernel<<<dim3(NB, NL / 16, ND / 16), dim3(32), 0, stream>>>(
        adj, text, Epart, G);
    tgc_out_wmma_kernel<<<dim3(NB * NL / 16, ND / 16), dim3(32), 0, stream>>>(
        G, Wm, bias, Spart, out);
}